// MultipassEmissionAbsorpsionRenderer_31507880084108
// MI455X (gfx1250) — compile-verified
//
#include <hip/hip_runtime.h>
#include <cstdint>

typedef __attribute__((ext_vector_type(4))) float v4f;

#define WAVES_PER_BLOCK 8
#define RPW 4                 // rays processed per wave (double-buffered)
#define NS 128                // samples per ray
#define BG_OPACITY 1.0e10f

// per-wave LDS buffer layout (floats): lengths[128] | densities[128] | features[384]
#define BUF_FLOATS 640
#define LEN_OFF 0
#define DEN_OFF 128
#define FEA_OFF 256

// ---- CDNA5 async global->LDS (ASYNCcnt) --------------------------------
__device__ __forceinline__ void async_b128(unsigned lds_byte, const float* g) {
  asm volatile("global_load_async_to_lds_b128 %0, %1, off"
               :: "v"(lds_byte), "v"(g)
               : "memory");
}
__device__ __forceinline__ void wait_async_le5() {
  asm volatile("s_wait_asynccnt 5" ::: "memory");
}
__device__ __forceinline__ void wait_async_0() {
  asm volatile("s_wait_asynccnt 0" ::: "memory");
}

// Issue 5 async b128 loads staging one ray (lengths, densities, features) into LDS.
__device__ __forceinline__ void prefetch_ray(const float* dens_g, const float* feat_g,
                                             const float* len_g, int ray,
                                             float* buf, int lane)
{
  const unsigned b = (unsigned)(uintptr_t)buf;   // low 32 bits of LDS flat addr = LDS offset
  const float* gl = len_g  + (size_t)ray * NS + lane * 4;
  const float* gd = dens_g + (size_t)ray * NS + lane * 4;
  const float* gf = feat_g + (size_t)ray * NS * 3 + lane * 12;
  async_b128(b + (LEN_OFF + lane * 4)  * 4u,       gl);
  async_b128(b + (DEN_OFF + lane * 4)  * 4u,       gd);
  async_b128(b + (FEA_OFF + lane * 12) * 4u,       gf);
  async_b128(b + (FEA_OFF + lane * 12) * 4u + 16u, gf + 4);
  async_b128(b + (FEA_OFF + lane * 12) * 4u + 32u, gf + 8);
}

__global__ __launch_bounds__(WAVES_PER_BLOCK * 32)
void ea_render(const float* __restrict__ dens_g,
               const float* __restrict__ feat_g,
               const float* __restrict__ len_g,
               const float* __restrict__ dir_g,
               float* __restrict__ out_feat,
               float* __restrict__ out_depth,
               float* __restrict__ out_opac,
               float* __restrict__ out_w,
               int total_rays)
{
  __shared__ __align__(16) float smem[WAVES_PER_BLOCK * 2 * BUF_FLOATS];
  const int lane = threadIdx.x & 31;
  const int wib  = threadIdx.x >> 5;
  const int wave = blockIdx.x * WAVES_PER_BLOCK + wib;

  float* buf0 = &smem[(wib * 2 + 0) * BUF_FLOATS];
  float* buf1 = &smem[(wib * 2 + 1) * BUF_FLOATS];

  const int ray0 = wave * RPW;
  if (ray0 >= total_rays) return;

  prefetch_ray(dens_g, feat_g, len_g, ray0, buf0, lane);

#pragma unroll
  for (int i = 0; i < RPW; ++i) {
    const int ray = ray0 + i;
    if (ray >= total_rays) break;
    float* cur = (i & 1) ? buf1 : buf0;
    float* nxt = (i & 1) ? buf0 : buf1;

    const bool pf = (i + 1 < RPW) && (ray + 1 < total_rays);
    if (pf) {
      prefetch_ray(dens_g, feat_g, len_g, ray + 1, nxt, lane); // overlap with compute
      wait_async_le5();   // next ray's 5 loads may remain in flight; current ray done
    } else {
      wait_async_0();
    }

    // ---- direction norm (wave-uniform broadcast loads) ----
    const float dx = dir_g[(size_t)ray * 3 + 0];
    const float dy = dir_g[(size_t)ray * 3 + 1];
    const float dz = dir_g[(size_t)ray * 3 + 2];
    const float dn = sqrtf(dx * dx + dy * dy + dz * dz);

    // ---- pull this lane's 4 samples out of LDS ----
    const v4f lv = *(const v4f*)(cur + LEN_OFF + lane * 4);
    const v4f dv = *(const v4f*)(cur + DEN_OFF + lane * 4);
    const float lnext = cur[LEN_OFF + ((lane * 4 + 4) & (NS - 1))]; // lane31 value unused
    const v4f fA = *(const v4f*)(cur + FEA_OFF + lane * 12 + 0);
    const v4f fB = *(const v4f*)(cur + FEA_OFF + lane * 12 + 4);
    const v4f fC = *(const v4f*)(cur + FEA_OFF + lane * 12 + 8);

    const float d0 = fmaxf(dv.x, 0.f), d1 = fmaxf(dv.y, 0.f);
    const float d2 = fmaxf(dv.z, 0.f), d3 = fmaxf(dv.w, 0.f);
    const float w0 = (lv.y - lv.x) * dn * d0;
    const float w1 = (lv.z - lv.y) * dn * d1;
    const float w2 = (lv.w - lv.z) * dn * d2;
    const float del3 = (lane == 31) ? BG_OPACITY : (lnext - lv.w);
    const float w3 = del3 * dn * d3;

    // ---- wave32 exclusive scan of lane sums (shift + Kogge-Stone) ----
    // Exclusive form avoids catastrophic cancellation with the 1e10 bg term.
    const float lsum = (w0 + w1) + (w2 + w3);
    float excl = __shfl_up(lsum, 1, 32);
    if (lane == 0) excl = 0.f;
#pragma unroll
    for (int off = 1; off < 32; off <<= 1) {
      const float t = __shfl_up(excl, off, 32);
      if (lane >= off) excl += t;
    }
    const float total = __shfl(excl + lsum, 31, 32);

    // per-sample exclusive cumsum and weights: (1-exp(-w)) * exp(-cumExcl)
    const float c0 = excl;
    const float c1 = c0 + w0;
    const float c2 = c1 + w1;
    const float c3 = c2 + w2;
    const float g0 = (1.f - __expf(-w0)) * __expf(-c0);
    const float g1 = (1.f - __expf(-w1)) * __expf(-c1);
    const float g2 = (1.f - __expf(-w2)) * __expf(-c2);
    const float g3 = (1.f - __expf(-w3)) * __expf(-c3);

    // ---- depth + feature dot products, butterfly reduce across wave ----
    float accD  = g0 * lv.x + g1 * lv.y + g2 * lv.z + g3 * lv.w;
    float accF0 = g0 * fA.x + g1 * fA.w + g2 * fB.z + g3 * fC.y;
    float accF1 = g0 * fA.y + g1 * fB.x + g2 * fB.w + g3 * fC.z;
    float accF2 = g0 * fA.z + g1 * fB.y + g2 * fC.x + g3 * fC.w;
#pragma unroll
    for (int off = 16; off > 0; off >>= 1) {
      accD  += __shfl_xor(accD,  off, 32);
      accF0 += __shfl_xor(accF0, off, 32);
      accF1 += __shfl_xor(accF1, off, 32);
      accF2 += __shfl_xor(accF2, off, 32);
    }

    // ---- stores: weights are a write-once stream -> nontemporal b128 ----
    v4f wv; wv.x = g0; wv.y = g1; wv.z = g2; wv.w = g3;
    __builtin_nontemporal_store(wv, (v4f*)(out_w + (size_t)ray * NS) + lane);

    if (lane == 0) {
      out_feat[(size_t)ray * 3 + 0] = accF0;   // BG_COLOR == 0 -> no bg blend term
      out_feat[(size_t)ray * 3 + 1] = accF1;
      out_feat[(size_t)ray * 3 + 2] = accF2;
      out_depth[ray] = accD;
      out_opac[ray]  = 1.f - __expf(-total);
    }
  }
}

extern "C" void kernel_launch(void* const* d_in, const int* in_sizes, int n_in,
                              void* d_out, int out_size, void* d_ws, size_t ws_size,
                              hipStream_t stream)
{
  (void)n_in; (void)out_size; (void)d_ws; (void)ws_size;
  const float* dens_g = (const float*)d_in[0];   // (B,R,N,1)
  const float* feat_g = (const float*)d_in[1];   // (B,R,N,3)
  const float* len_g  = (const float*)d_in[2];   // (B,R,N)
  const float* dir_g  = (const float*)d_in[3];   // (B,R,3)

  const int total_rays = in_sizes[3] / 3;        // B*R

  float* out       = (float*)d_out;              // features | depths | opacities | weights
  float* out_feat  = out;
  float* out_depth = out_feat + (size_t)total_rays * 3;
  float* out_opac  = out_depth + total_rays;
  float* out_w     = out_opac + total_rays;

  const int waves  = (total_rays + RPW - 1) / RPW;
  const int blocks = (waves + WAVES_PER_BLOCK - 1) / WAVES_PER_BLOCK;
  ea_render<<<blocks, WAVES_PER_BLOCK * 32, 0, stream>>>(
      dens_g, feat_g, len_g, dir_g, out_feat, out_depth, out_opac, out_w, total_rays);
}